// MyHierarchicalDense_66159676228327
// MI455X (gfx1250) — compile-verified
//
#include <hip/hip_runtime.h>

// Problem constants (from reference): B=256, G=64, U=1024, P=1024, f32.
#define BN 256
#define GN 64
#define UN 1024
#define PN 1024

#define TU 128            // U columns per block
#define KB 64             // K panel staged in LDS
#define LDW 68            // padded LDS row stride (floats): bank-conflict-free
#define NUT (UN / TU)     // 8 U-tiles
#define NPANEL (PN / KB)  // 16 K panels
#define PER_PANEL 9       // async ops per wave per panel (8 W + 1 X)

typedef __attribute__((ext_vector_type(2))) float v2f;
typedef __attribute__((ext_vector_type(8))) float v8f;
typedef int v4i __attribute__((vector_size(16)));   // async builtin pointee type

#if __has_builtin(__builtin_amdgcn_global_load_async_to_lds_b128)
#define HAVE_ASYNC 1
#else
#define HAVE_ASYNC 0
#endif

template <int N>
__device__ __forceinline__ void wait_async() {
#if __has_builtin(__builtin_amdgcn_s_wait_asynccnt)
    __builtin_amdgcn_s_wait_asynccnt(N);
#else
    asm volatile("s_wait_asynccnt %0" :: "i"(N) : "memory");
#endif
}

#if HAVE_ASYNC
// Issue async global->LDS copies for one K panel (no waits here).
// W gets a non-temporal hint (streamed exactly once device-wide).
__device__ __forceinline__ void stage_panel_async(
    const float* __restrict__ Wg, const float* __restrict__ xrow,
    float* WsBuf, float* XsBuf, int uBase, int p0, int t)
{
    // W panel: 128 rows x 64 f32 = 2048 b128 copies, 8 per thread.
    #pragma unroll
    for (int i = 0; i < 8; ++i) {
        int idx4 = t + i * 256;            // 0..2047
        int row  = idx4 >> 4;              // 0..127
        int c4   = (idx4 & 15) << 2;       // 0,4,...,60
        const float* gp = Wg + (size_t)(uBase + row) * PN + p0 + c4;
        float*       lp = &WsBuf[row * LDW + c4];
        __builtin_amdgcn_global_load_async_to_lds_b128(
            (v4i*)gp, (v4i*)lp, 0, /*cpol: TH=NT*/1);
    }
    // X panel: one b128 per thread from the pre-resolved gather pointer.
    {
        int row = t >> 4;                  // 0..15
        int c4  = (t & 15) << 2;
        float*  lp = &XsBuf[row * LDW + c4];
        __builtin_amdgcn_global_load_async_to_lds_b128(
            (v4i*)(xrow + p0), (v4i*)lp, 0, 0);
    }
}
#endif

// 16 x v_wmma_f32_16x16x4_f32 over one K panel, two independent acc chains,
// A/B fragments rotated one step ahead so LDS loads overlap the WMMAs.
__device__ __forceinline__ void compute_panel(
    const float* WsBuf, const float* XsBuf,
    int lane, int wave, v8f& acc0, v8f& acc1)
{
    const int nrow = wave * 16 + (lane & 15);
    const int koff = (lane >> 4) << 1;          // 0 or 2
    const float* xb = &XsBuf[(lane & 15) * LDW + koff];
    const float* wb = &WsBuf[nrow * LDW + koff];

    v2f a0 = *(const v2f*)(xb);     v2f a1 = *(const v2f*)(xb + 4);
    v2f b0 = *(const v2f*)(wb);     v2f b1 = *(const v2f*)(wb + 4);
    #pragma unroll
    for (int k = 0; k < KB; k += 8) {
        v2f na0 = a0, na1 = a1, nb0 = b0, nb1 = b1;
        if (k + 8 < KB) {
            na0 = *(const v2f*)(xb + k + 8);  na1 = *(const v2f*)(xb + k + 12);
            nb0 = *(const v2f*)(wb + k + 8);  nb1 = *(const v2f*)(wb + k + 12);
        }
        acc0 = __builtin_amdgcn_wmma_f32_16x16x4_f32(
                false, a0, false, b0, (short)0, acc0, false, false);
        acc1 = __builtin_amdgcn_wmma_f32_16x16x4_f32(
                false, a1, false, b1, (short)0, acc1, false, false);
        a0 = na0; a1 = na1; b0 = nb0; b1 = nb1;
    }
}

__global__ __launch_bounds__(256) void
hd_wmma_kernel(const float* __restrict__ x,      // [B, P]
               const int*   __restrict__ gid,    // [B]
               const float* __restrict__ W,      // [G, U, P]
               const float* __restrict__ bias,   // [G, U]
               float*       __restrict__ out)    // [B, U]
{
    __shared__ float Ws[3][TU * LDW];
    __shared__ float Xs[3][16 * LDW];
    __shared__ int   lists[BN];
    __shared__ int   scnt;

    const int g     = blockIdx.x / NUT;
    const int uBase = (blockIdx.x % NUT) * TU;
    const int t     = threadIdx.x;
    const int lane  = t & 31;
    const int wave  = t >> 5;

    // ---- build this group's example list (order-independent result) ----
    if (t == 0) scnt = 0;
    __syncthreads();
    if (gid[t] == g) {
        int pos = atomicAdd(&scnt, 1);
        lists[pos] = t;
    }
    __syncthreads();
    const int cnt = scnt;
    if (cnt == 0) return;               // uniform exit: group unused

    const float* Wg = W + (size_t)g * UN * PN;

    // bias for this wave's N column (C layout: N = lane % 16 in every VGPR)
    const float bv = bias[g * UN + uBase + wave * 16 + (lane & 15)];

    const int numMT = (cnt + 15) >> 4;  // usually 1 (avg 4 rows/group)
    for (int mt = 0; mt < numMT; ++mt) {
        const int mBase = mt << 4;

        // Pre-resolve this thread's X gather pointer for the whole M-tile.
        // Pad rows by clamping the index (padded rows only feed outputs we
        // never store, so duplicated data is harmless and keeps EXEC full).
        int mrow = mBase + (t >> 4);
        int src  = lists[mrow < cnt ? mrow : cnt - 1];
        const float* xrow = x + (size_t)src * PN + ((t & 15) << 2);

        v8f acc0, acc1;
        #pragma unroll
        for (int j = 0; j < 8; ++j) { acc0[j] = bv; acc1[j] = 0.f; }

#if HAVE_ASYNC
        // ---- triple-buffered async pipeline, 2 panels in flight ----
        stage_panel_async(Wg, xrow, Ws[0], Xs[0], uBase, 0, t);
        stage_panel_async(Wg, xrow, Ws[1], Xs[1], uBase, KB, t);
        wait_async<PER_PANEL>();        // panel 0 landed; panel 1 in flight
        __syncthreads();
        for (int i = 0; i < NPANEL; ++i) {
            if (i + 2 < NPANEL)
                stage_panel_async(Wg, xrow, Ws[(i + 2) % 3], Xs[(i + 2) % 3],
                                  uBase, (i + 2) * KB, t);
            compute_panel(Ws[i % 3], Xs[i % 3], lane, wave, acc0, acc1);
            if (i + 2 < NPANEL)
                wait_async<PER_PANEL>();   // in-order: panel i+1 landed
            else
                wait_async<0>();           // tail: drain everything
            __syncthreads();               // everyone done reading panel i
        }
#else
        // ---- fallback: synchronous staging, batched loads for MLP ----
        for (int p0 = 0; p0 < PN; p0 += KB) {
            __syncthreads();
            float4 wtmp[8];
            #pragma unroll
            for (int i = 0; i < 8; ++i) {
                int idx4 = t + i * 256;
                int row  = idx4 >> 4;
                int c4   = (idx4 & 15) << 2;
                wtmp[i] = *(const float4*)(Wg + (size_t)(uBase + row) * PN + p0 + c4);
            }
            float4 xtmp = *(const float4*)(xrow + p0);
            #pragma unroll
            for (int i = 0; i < 8; ++i) {
                int idx4 = t + i * 256;
                int row  = idx4 >> 4;
                int c4   = (idx4 & 15) << 2;
                *(float4*)(&Ws[0][row * LDW + c4]) = wtmp[i];
            }
            *(float4*)(&Xs[0][(t >> 4) * LDW + ((t & 15) << 2)]) = xtmp;
            __syncthreads();
            compute_panel(Ws[0], Xs[0], lane, wave, acc0, acc1);
        }
#endif

        // ---- epilogue: combine chains, relu, scatter rows by example id ----
        const int n = uBase + wave * 16 + (lane & 15);
        #pragma unroll
        for (int j = 0; j < 8; ++j) {
            int m = mBase + j + ((lane >> 4) << 3);   // M = j + 8*(lane>=16)
            if (m < cnt) {
                float v = acc0[j] + acc1[j];
                out[(size_t)lists[m] * UN + n] = v > 0.f ? v : 0.f;
            }
        }
    }
}

extern "C" void kernel_launch(void* const* d_in, const int* in_sizes, int n_in,
                              void* d_out, int out_size, void* d_ws, size_t ws_size,
                              hipStream_t stream) {
    const float* x    = (const float*)d_in[0];
    const int*   gid  = (const int*)  d_in[1];
    const float* W    = (const float*)d_in[2];
    const float* bias = (const float*)d_in[3];
    float*       out  = (float*)d_out;

    hd_wmma_kernel<<<dim3(GN * NUT), dim3(256), 0, stream>>>(x, gid, W, bias, out);
}